// PositionAttention_41334765257058
// MI455X (gfx1250) — compile-verified
//
#include <hip/hip_runtime.h>
#include <hip/hip_bf16.h>

#define BB 16
#define CC 256
#define HH 96
#define WW 96
#define HW 9216
#define NS 50
#define SP 52          // padded token count (13 * 4 for WMMA f32 K=4 steps)
#define NBLK 36        // HW / 256
#define ZPART (NBLK*8) // partial sums per (b,s): 36 blocks * 8 waves = 288

typedef __attribute__((ext_vector_type(2))) float v2f;
typedef __attribute__((ext_vector_type(8))) float v8f;

// ---------------------------------------------------------------------------
// K1: multi-scale pooling. One block per (b,c) plane. 16x16 region sums first,
// then all 50 pools (k=1,2,3,6) are combinations of those 36 region sums.
__global__ __launch_bounds__(256) void pool_kernel(const float* __restrict__ x,
                                                   float* __restrict__ F) {
  const int bc = blockIdx.x;                 // b*C + c
  const float* xp = x + (size_t)bc * HW;
  __shared__ float rowsum[576];
  __shared__ float regsum[36];
  const int tid = threadIdx.x;
  for (int pass = 0; pass < 3; ++pass) {
    int idx = tid + pass * 256;
    if (idx < 576) {
      int r = idx >> 4, row = idx & 15;
      int ri = r / 6, rj = r % 6;
      const float4* p4 = reinterpret_cast<const float4*>(xp + (ri*16+row)*WW + rj*16);
      float4 a = p4[0], b = p4[1], c = p4[2], d = p4[3];
      rowsum[idx] = (a.x+a.y+a.z+a.w) + (b.x+b.y+b.z+b.w) +
                    (c.x+c.y+c.z+c.w) + (d.x+d.y+d.z+d.w);
    }
  }
  __syncthreads();
  if (tid < 36) {
    float s = 0.f;
    for (int row = 0; row < 16; ++row) s += rowsum[tid*16 + row];
    regsum[tid] = s;
  }
  __syncthreads();
  if (tid < SP) {
    float v = 0.f;
    if (tid == 0) {                                 // k=1
      for (int r = 0; r < 36; ++r) v += regsum[r];
      v *= (1.f/9216.f);
    } else if (tid < 5) {                           // k=2 : 3x3 regions each
      int q = tid-1, i = q>>1, j = q&1;
      for (int a = 0; a < 3; ++a) for (int b2 = 0; b2 < 3; ++b2)
        v += regsum[(3*i+a)*6 + 3*j+b2];
      v *= (1.f/2304.f);
    } else if (tid < 14) {                          // k=3 : 2x2 regions each
      int q = tid-5, i = q/3, j = q%3;
      for (int a = 0; a < 2; ++a) for (int b2 = 0; b2 < 2; ++b2)
        v += regsum[(2*i+a)*6 + 2*j+b2];
      v *= (1.f/1024.f);
    } else if (tid < 50) {                          // k=6
      v = regsum[tid-14] * (1.f/256.f);
    }                                               // tid 50,51 -> 0 pad
    F[(size_t)bc * SP + tid] = v;
  }
}

// ---------------------------------------------------------------------------
// K2: Bm = relu(w_b . x) over channels, one pixel per thread (coalesced in n).
__global__ __launch_bounds__(256) void bm_kernel(const float* __restrict__ x,
                                                 const float* __restrict__ wb,
                                                 float* __restrict__ Bm) {
  __shared__ float w[CC];
  const int b = blockIdx.y;
  const int n = blockIdx.x*256 + threadIdx.x;
  w[threadIdx.x] = wb[threadIdx.x];
  __syncthreads();
  const float* xb = x + (size_t)b*CC*HW + n;
  float acc = 0.f;
  for (int c = 0; c < CC; ++c) acc += w[c] * xb[(size_t)c*HW];
  Bm[b*HW + n] = fmaxf(acc, 0.f);
}

// ---------------------------------------------------------------------------
// K3: Cf = relu(w_fc . F)  -> [B, SP]
__global__ __launch_bounds__(64) void cf_kernel(const float* __restrict__ wfc,
                                                const float* __restrict__ F,
                                                float* __restrict__ Cf) {
  const int b = blockIdx.x;
  const int s = threadIdx.x;
  if (s < SP) {
    float acc = 0.f;
    if (s < NS) {
      for (int c = 0; c < CC; ++c) acc += wfc[c] * F[((size_t)b*CC + c)*SP + s];
      acc = fmaxf(acc, 0.f);
    }
    Cf[b*SP + s] = acc;
  }
}

// ---------------------------------------------------------------------------
// K4/K5: Out[b] = (relu?)(W @ Bmat[b]) : M=256, N=52(pad64), K=256 f32 WMMA.
// One wave per 16x16 tile; 64 K-steps of V_WMMA_F32_16X16X4_F32.
__global__ __launch_bounds__(32) void gemm_wmma(const float* __restrict__ W,
                                                const float* __restrict__ Bmat,
                                                float* __restrict__ Out, int relu) {
  const int tile = blockIdx.x;      // (C/16)*4 tiles
  const int b    = blockIdx.y;
  const int o0 = (tile >> 2) * 16;
  const int s0 = (tile & 3) * 16;
  const int lane  = threadIdx.x & 31;
  const int lhalf = lane >> 4;      // 0: lanes 0-15, 1: lanes 16-31
  const int m     = lane & 15;
  v8f acc = {};
  for (int k = 0; k < 64; ++k) {
    const int kk = 4*k + 2*lhalf;   // A: K=kk,kk+1 per ISA 16x4 f32 layout
    v2f a, bv;
    a[0] = W[(o0 + m)*CC + kk];
    a[1] = W[(o0 + m)*CC + kk + 1];
    bv[0] = Bmat[((size_t)b*CC + kk    )*SP + s0 + m];
    bv[1] = Bmat[((size_t)b*CC + kk + 1)*SP + s0 + m];
    acc = __builtin_amdgcn_wmma_f32_16x16x4_f32(false, a, false, bv,
                                                (short)0, acc, false, false);
  }
#pragma unroll
  for (int r = 0; r < 8; ++r) {
    const int o = o0 + r + lhalf*8;
    const int s = s0 + m;
    if (s < SP) {
      float v = acc[r];
      if (relu) v = fmaxf(v, 0.f);
      Out[((size_t)b*CC + o)*SP + s] = v;
    }
  }
}

// ---------------------------------------------------------------------------
// K6: maxBm per batch (Bm >= 0 so 0 is a safe floor).
__global__ __launch_bounds__(256) void maxbm_kernel(const float* __restrict__ Bm,
                                                    float* __restrict__ Mx) {
  const int b = blockIdx.x;
  float m = 0.f;
  for (int i = threadIdx.x; i < HW; i += 256) m = fmaxf(m, Bm[b*HW + i]);
  for (int off = 16; off > 0; off >>= 1) m = fmaxf(m, __shfl_xor(m, off, 32));
  __shared__ float wmax[8];
  const int lane = threadIdx.x & 31, w = threadIdx.x >> 5;
  if (lane == 0) wmax[w] = m;
  __syncthreads();
  if (threadIdx.x == 0) {
    float r = wmax[0];
    for (int i = 1; i < 8; ++i) r = fmaxf(r, wmax[i]);
    Mx[b] = r;
  }
}

// ---------------------------------------------------------------------------
// K7: e[b,s,n] = exp(Cf[s]*(Bm[n]-maxBm)) table + deterministic wave-level
//     partial sums for the softmax denominator (no float atomics).
__global__ __launch_bounds__(256) void expz_kernel(const float* __restrict__ Bm,
                                                   const float* __restrict__ Cf,
                                                   const float* __restrict__ Mx,
                                                   float* __restrict__ Eb,
                                                   float* __restrict__ Zpart) {
  const int b = blockIdx.y, nb = blockIdx.x;
  const int n = nb*256 + threadIdx.x;
  const int lane = threadIdx.x & 31, w = threadIdx.x >> 5;
  const float bmv = Bm[b*HW + n] - Mx[b];
  float* ebase = Eb + (size_t)b*SP*HW + n;
  for (int s = 0; s < NS; ++s) {
    float v = __expf(Cf[b*SP + s] * bmv);   // logit - max <= 0: no overflow
    ebase[(size_t)s*HW] = v;
    float r = v;
    for (int off = 16; off > 0; off >>= 1) r += __shfl_xor(r, off, 32);
    if (lane == 0) Zpart[((size_t)b*NS + s)*ZPART + nb*8 + w] = r;
  }
  ebase[(size_t)NS*HW]     = 0.f;           // K-pad rows 50,51
  ebase[(size_t)(NS+1)*HW] = 0.f;
}

// K7b: fixed-order reduction of Z partials (deterministic).
__global__ __launch_bounds__(64) void zred_kernel(const float* __restrict__ Zpart,
                                                  float* __restrict__ Z) {
  const int b = blockIdx.x;
  const int s = threadIdx.x;
  if (s < SP) {
    float acc = 0.f;
    if (s < NS) {
      const float* p = Zpart + ((size_t)b*NS + s)*ZPART;
      for (int i = 0; i < ZPART; ++i) acc += p[i];
    }
    Z[b*SP + s] = acc;
  }
}

// ---------------------------------------------------------------------------
// K8: H = G / Z (softmax denominator folded into the left GEMM operand).
__global__ __launch_bounds__(256) void hscale_kernel(float* __restrict__ G,
                                                     const float* __restrict__ Z) {
  const int idx = blockIdx.x*256 + threadIdx.x;   // B*C*SP elements
  const int s = idx % SP;
  const int b = (idx / SP) / CC;
  G[idx] = (s < NS) ? G[idx] / Z[b*SP + s] : 0.f;
}

// ---------------------------------------------------------------------------
// K9 (main): out[b,c,n] = sum_s H[b,c,s] * e[b,s,n] + x[b,c,n].
// Per-batch GEMM M=256, N=9216, K=52 via f32 WMMA. 8 waves/block; H-tile
// staged in LDS; e rows streamed from the L2-resident table; NT stores.
__global__ __launch_bounds__(256) void attn_main_kernel(const float* __restrict__ x,
                                                        const float* __restrict__ Hm,
                                                        const float* __restrict__ Eb,
                                                        float* __restrict__ out) {
  const int b = blockIdx.z;
  const int c0 = blockIdx.y * 16;
  const int tid = threadIdx.x;
  const int wave = tid >> 5, lane = tid & 31;
  const int lhalf = lane >> 4;
  const int m = lane & 15;
  const int n0 = blockIdx.x * 128 + wave * 16;

  __shared__ float Hs[16][SP];
  for (int idx = tid; idx < 16*SP; idx += 256) {
    int row = idx / SP, s = idx - row*SP;
    Hs[row][s] = Hm[((size_t)b*CC + c0 + row)*SP + s];
  }
  __syncthreads();

  const float* ebase = Eb + (size_t)b*SP*HW + n0 + m;
  v8f acc = {};
  for (int k = 0; k < 13; ++k) {
    const int sb = 4*k + 2*lhalf;
    v2f a, bv;
    a[0] = Hs[m][sb];
    a[1] = Hs[m][sb+1];
    bv[0] = ebase[(size_t)sb*HW];
    bv[1] = ebase[(size_t)(sb+1)*HW];
    acc = __builtin_amdgcn_wmma_f32_16x16x4_f32(false, a, false, bv,
                                                (short)0, acc, false, false);
  }

  const size_t obase = ((size_t)b*CC + c0 + lhalf*8)*HW + n0 + m;
#pragma unroll
  for (int r = 0; r < 8; ++r) {
    const size_t idx = obase + (size_t)r*HW;
    __builtin_nontemporal_store(acc[r] + x[idx], &out[idx]);  // keep x in L2
  }
}

// ---------------------------------------------------------------------------
extern "C" void kernel_launch(void* const* d_in, const int* in_sizes, int n_in,
                              void* d_out, int out_size, void* d_ws, size_t ws_size,
                              hipStream_t stream) {
  (void)in_sizes; (void)n_in; (void)out_size; (void)ws_size;
  const float* x       = (const float*)d_in[0];
  const float* w_fc    = (const float*)d_in[1];
  const float* w_fd    = (const float*)d_in[2];
  const float* w_b     = (const float*)d_in[3];
  const float* w_alpha = (const float*)d_in[4];
  float* out = (float*)d_out;
  float* ws  = (float*)d_ws;

  // workspace layout (floats)
  float* F     = ws;                     // 16*256*52   = 212,992
  float* Dm    = ws + 212992;            // 212,992
  float* G     = ws + 425984;            // 212,992  (becomes H in place)
  float* Eb    = ws + 638976;            // 16*52*9216  = 7,667,712
  float* Bm    = ws + 8306688;           // 147,456
  float* Cf    = ws + 8454144;           // 832
  float* Z     = ws + 8454976;           // 832
  float* Mx    = ws + 8455808;           // 16
  float* Zpart = ws + 8455824;           // 16*50*288   = 230,400  (end 8,686,224)

  pool_kernel<<<BB*CC, 256, 0, stream>>>(x, F);
  bm_kernel<<<dim3(NBLK, BB), 256, 0, stream>>>(x, w_b, Bm);
  cf_kernel<<<BB, 64, 0, stream>>>(w_fc, F, Cf);
  gemm_wmma<<<dim3(64, BB), 32, 0, stream>>>(w_fd, F, Dm, 1);     // D = relu(w_fd@F)
  gemm_wmma<<<dim3(64, BB), 32, 0, stream>>>(w_alpha, Dm, G, 0);  // G = w_alpha@D
  maxbm_kernel<<<BB, 256, 0, stream>>>(Bm, Mx);
  expz_kernel<<<dim3(NBLK, BB), 256, 0, stream>>>(Bm, Cf, Mx, Eb, Zpart);
  zred_kernel<<<BB, 64, 0, stream>>>(Zpart, Z);
  hscale_kernel<<<(BB*CC*SP)/256, 256, 0, stream>>>(G, Z);
  attn_main_kernel<<<dim3(HW/128, CC/16, BB), 256, 0, stream>>>(x, G, Eb, out);
}